// SparsePropMaxPool_6004364280201
// MI455X (gfx1250) — compile-verified
//
#include <hip/hip_runtime.h>

// Output layout (flat, reference return order):
//   props_h      : (64, 153, 512)      -> offset 0,            5,013,504 floats
//   ori_map_h    : (64, 512, 64, 64)   -> offset 5,013,504,  134,217,728 floats
//   ori_map_mask : (64, 1, 64, 64)     -> offset 139,231,232,    262,144 floats
#define PROPS_OFF 0ULL
#define ORI_OFF   5013504ULL
#define MASK_OFF  139231232ULL

typedef int v4i __attribute__((vector_size(16)));
typedef __attribute__((address_space(1))) v4i as1_v4i;   // global-AS 4xi32
typedef __attribute__((address_space(3))) v4i as3_v4i;   // LDS-AS    4xi32
typedef __attribute__((address_space(3))) char as3_char;

// ---- CDNA5 async LDS->global store (ASYNCcnt-tracked DMA path) ----
// cpol = 1 -> TH_NT: 512MB write stream > 192MB L2, keep it non-temporal.
__device__ __forceinline__ void async_store_b128(void* g, void* l) {
#if __has_builtin(__builtin_amdgcn_global_store_async_from_lds_b128)
    __builtin_amdgcn_global_store_async_from_lds_b128((as1_v4i*)g, (as3_v4i*)l,
                                                      /*imm offset*/0, /*cpol*/1);
#else
    unsigned long long ga = (unsigned long long)g;
    unsigned la = (unsigned)(unsigned long long)(as3_char*)l;  // raw 32-bit LDS byte addr
    asm volatile("global_store_async_from_lds_b128 %0, %1, off th:TH_STORE_NT"
                 :: "v"(ga), "v"(la) : "memory");
#endif
}

__device__ __forceinline__ void wait_async0() {
#if __has_builtin(__builtin_amdgcn_s_wait_asynccnt)
    __builtin_amdgcn_s_wait_asynccnt(0);
#else
    asm volatile("s_wait_asynccnt 0" ::: "memory");
#endif
}

// Closed-form props table (153 entries) from _build_props():
//   [0,57):   (s, s+7),  s=0..56
//   [57,106): (s, s+15), s=0..48
//   [106,127):(2k, 2k+23), k=0..20
//   [127,144):(2k, 2k+31), k=0..16
//   [144,149):(4k, 4k+47), k=0..4
//   [149,152):(4k, 4k+55), k=0..2
//   152:      (0, 63)
__device__ __forceinline__ void prop_se(int p, int& s, int& e) {
    if (p < 57)       { s = p;              e = s + 7;  }
    else if (p < 106) { s = p - 57;         e = s + 15; }
    else if (p < 127) { s = (p - 106) * 2;  e = s + 23; }
    else if (p < 144) { s = (p - 127) * 2;  e = s + 31; }
    else if (p < 149) { s = (p - 144) * 4;  e = s + 47; }
    else if (p < 152) { s = (p - 149) * 4;  e = s + 55; }
    else              { s = 0;              e = 63;     }
}

// One block (64 threads = 2 wave32) per (b,h) row.
extern "C" __global__ void __launch_bounds__(64)
pool_scatter_kernel(const float* __restrict__ x, float* __restrict__ out) {
    __shared__ __align__(16) float xs[64];
    __shared__ __align__(16) float tile[64 * 64];   // 16 KB of the 320 KB WGP LDS

    const int t   = threadIdx.x;      // 0..63
    const int row = blockIdx.x;       // b*512 + h

    // Zero the tile with b128 LDS stores.
    float4* t4 = (float4*)tile;
    const float4 z = make_float4(0.f, 0.f, 0.f, 0.f);
#pragma unroll
    for (int i = 0; i < 16; ++i) t4[t + i * 64] = z;

    xs[t] = x[(size_t)row * 64 + t];
    __syncthreads();

    // Thread t owns map row s = t: running inclusive-window max.
    {
        const int s = t;
        float m = xs[s];
        float* trow = tile + s * 64;
#pragma unroll
        for (int w = 1; w <= 16; ++w) {                 // widths 1..16, any s
            const int e = s + w - 1;
            if (e < 64) { m = fmaxf(m, xs[e]); trow[e] = m; }
        }
        if ((s & 1) == 0) {
#pragma unroll
            for (int w = 18; w <= 32; w += 2) {         // widths 18..32, even s
                const int e = s + w - 1;
                if (e < 64) { m = fmaxf(m, fmaxf(xs[e - 1], xs[e])); trow[e] = m; }
            }
        }
        if ((s & 3) == 0) {
#pragma unroll
            for (int w = 36; w <= 64; w += 4) {         // widths 36..64, s%4==0
                const int e = s + w - 1;
                if (e < 64) {
                    m = fmaxf(m, fmaxf(fmaxf(xs[e - 3], xs[e - 2]),
                                       fmaxf(xs[e - 1], xs[e])));
                    trow[e] = m;
                }
            }
        }
    }
    __syncthreads();

    // props_h[b][p][h] = tile[p0][p1-1]
    const int b = row >> 9;
    const int h = row & 511;
    for (int p = t; p < 153; p += 64) {
        int s0, e0;
        prop_se(p, s0, e0);
        out[PROPS_OFF + ((size_t)(b * 153 + p)) * 512 + h] = tile[s0 * 64 + e0];
    }

    // Stream the 16 KB tile to ori_map_h[row] via async LDS->global DMA (NT).
    char* gbase = (char*)(out + ORI_OFF + (size_t)row * 4096);
#pragma unroll
    for (int i = 0; i < 16; ++i) {
        const int c = t + i * 64;                       // float4 chunk index
        async_store_b128(gbase + (size_t)c * 16, (char*)tile + c * 16);
    }
    wait_async0();
}

// ori_map_mask: (64,1,64,64) -> 1.0 at valid (s,e) positions, else 0.
extern "C" __global__ void __launch_bounds__(256)
mask_kernel(float* __restrict__ out) {
    const int idx = blockIdx.x * 256 + threadIdx.x;     // 0..262143
    const int r = idx & 4095;
    const int s = r >> 6;
    const int e = r & 63;
    const int d = e - s;
    const bool v = (d >= 0) &&
        ((d <= 15) ||
         (((s & 1) == 0) && (d & 1) && d >= 17 && d <= 31) ||
         (((s & 3) == 0) && ((d & 3) == 3) && d >= 35));
    out[MASK_OFF + idx] = v ? 1.0f : 0.0f;
}

extern "C" void kernel_launch(void* const* d_in, const int* in_sizes, int n_in,
                              void* d_out, int out_size, void* d_ws, size_t ws_size,
                              hipStream_t stream) {
    (void)in_sizes; (void)n_in; (void)out_size; (void)d_ws; (void)ws_size;
    const float* x = (const float*)d_in[0];   // (64,512,64) f32
    // d_in[1] (props) is reproduced in closed form on-device; d_in[2] unused.
    float* out = (float*)d_out;

    pool_scatter_kernel<<<dim3(64 * 512), dim3(64), 0, stream>>>(x, out);
    mask_kernel<<<dim3(1024), dim3(256), 0, stream>>>(out);
}